// TransformerBlock_40587440947732
// MI455X (gfx1250) — compile-verified
//
#include <hip/hip_runtime.h>
#include <math.h>

// ---------------------------------------------------------------------------
// Transformer block on MI455X (gfx1250), all GEMMs via native fp32 WMMA
// V_WMMA_F32_16X16X4_F32  (D = A(16x4) * B(4x16) + C(16x16)), wave32.
//
// Fragment layouts (CDNA5 ISA 7.12.2):
//   A 16x4 f32 : lane L holds row M=L&15, K = 2*(L>>4) + {vgpr0,vgpr1}
//   B 4x16 f32 : lane L holds col N=L&15, K = 2*(L>>4) + {vgpr0,vgpr1}
//   C 16x16 f32: vgpr i, lane L -> row M = i + 8*(L>>4), col N = L&15
// A and B fragments are single contiguous 8-byte loads when the source is
// K-contiguous row-major (true for activations and all [N,K] weights).
// ---------------------------------------------------------------------------

typedef float v2f __attribute__((ext_vector_type(2)));
typedef float v8f __attribute__((ext_vector_type(8)));

__device__ __forceinline__ v8f wmma_f32(v2f a, v2f b, v8f c) {
  // 8 args: (neg_a, A, neg_b, B, c_mod, C, reuse_a, reuse_b)
  return __builtin_amdgcn_wmma_f32_16x16x4_f32(false, a, false, b, (short)0, c,
                                               false, false);
}

#define FLAG_BIAS 1
#define FLAG_RES 2
#define FLAG_RELU 4

// Fragment bundle for one K-step of the 32x64 wave tile (2 A rows, 4 W tiles)
struct Frags {
  v2f a0, a1, b0, b1, b2, b3;
};

__device__ __forceinline__ void load_frags(Frags& f, const float* pa0,
                                           const float* pa1, const float* pw0,
                                           const float* pw1, const float* pw2,
                                           const float* pw3, int off) {
  f.a0 = *(const v2f*)(pa0 + off);
  f.a1 = *(const v2f*)(pa1 + off);
  f.b0 = *(const v2f*)(pw0 + off);
  f.b1 = *(const v2f*)(pw1 + off);
  f.b2 = *(const v2f*)(pw2 + off);
  f.b3 = *(const v2f*)(pw3 + off);
}

__device__ __forceinline__ void mma_frags(v8f acc[8], const Frags& f) {
  acc[0] = wmma_f32(f.a0, f.b0, acc[0]);
  acc[1] = wmma_f32(f.a0, f.b1, acc[1]);
  acc[2] = wmma_f32(f.a0, f.b2, acc[2]);
  acc[3] = wmma_f32(f.a0, f.b3, acc[3]);
  acc[4] = wmma_f32(f.a1, f.b0, acc[4]);
  acc[5] = wmma_f32(f.a1, f.b1, acc[5]);
  acc[6] = wmma_f32(f.a1, f.b2, acc[6]);
  acc[7] = wmma_f32(f.a1, f.b3, acc[7]);
}

// ---------------------------------------------------------------------------
// C[M,N] = A[M,K] @ W[N,K]^T (+bias[N]) (+res[M,N]) (+relu)
// Block = 128 threads = 4 waves. Each wave: 32 rows x 64 cols of C
// (2 M-tiles x 4 N-tiles; 6 b64 loads per 8 WMMAs). 2-stage register
// pipeline over K so loads overlap WMMAs. grid = (M/32, N/256).
// Requires K % 8 == 0 (K here is 256/1024/2048).
// ---------------------------------------------------------------------------
__global__ __launch_bounds__(128) void gemm_wmma_f32_kernel(
    const float* __restrict__ A, const float* __restrict__ W,
    const float* __restrict__ bias, const float* __restrict__ res,
    float* __restrict__ C, int M, int N, int K, int flags) {
  const int lane = threadIdx.x & 31;
  const int wave = threadIdx.x >> 5;
  const int r16 = lane & 15;
  const int hi = lane >> 4;
  const int klo = hi << 1;

  const int mrow = blockIdx.x * 32;
  const int nb = blockIdx.y * 256 + wave * 64;

  const float* pa0 = A + (size_t)(mrow + r16) * K + klo;
  const float* pa1 = A + (size_t)(mrow + 16 + r16) * K + klo;
  const float* pw0 = W + (size_t)(nb + 0 * 16 + r16) * K + klo;
  const float* pw1 = W + (size_t)(nb + 1 * 16 + r16) * K + klo;
  const float* pw2 = W + (size_t)(nb + 2 * 16 + r16) * K + klo;
  const float* pw3 = W + (size_t)(nb + 3 * 16 + r16) * K + klo;

  v8f acc[8];
  {
    v8f zero = {0, 0, 0, 0, 0, 0, 0, 0};
#pragma unroll
    for (int i = 0; i < 8; ++i) acc[i] = zero;
  }

  Frags f0, f1;
  load_frags(f0, pa0, pa1, pw0, pw1, pw2, pw3, 0);  // step k=0
  int k = 0;
  for (; k < K - 8; k += 8) {
    load_frags(f1, pa0, pa1, pw0, pw1, pw2, pw3, 4);  // step k+4
    mma_frags(acc, f0);                               // step k
    load_frags(f0, pa0, pa1, pw0, pw1, pw2, pw3, 8);  // step k+8
    mma_frags(acc, f1);                               // step k+4
    pa0 += 8;
    pa1 += 8;
    pw0 += 8;
    pw1 += 8;
    pw2 += 8;
    pw3 += 8;
  }
  // k == K-8: f0 holds step K-8
  load_frags(f1, pa0, pa1, pw0, pw1, pw2, pw3, 4);  // step K-4
  mma_frags(acc, f0);
  mma_frags(acc, f1);

#pragma unroll
  for (int mm = 0; mm < 2; ++mm) {
#pragma unroll
    for (int t = 0; t < 4; ++t) {
#pragma unroll
      for (int i = 0; i < 8; ++i) {
        const int row = mrow + mm * 16 + i + 8 * hi;
        const int col = nb + t * 16 + r16;
        float val = acc[mm * 4 + t][i];
        if (flags & FLAG_BIAS) val += bias[col];
        if (flags & FLAG_RES) val += res[(size_t)row * N + col];
        if (flags & FLAG_RELU) val = fmaxf(val, 0.0f);
        C[(size_t)row * N + col] = val;
      }
    }
  }
}

// ---------------------------------------------------------------------------
// Flash attention, fp32 WMMA. One wave (32 threads) per block.
// grid = (64 q-tiles, 64 b*h). q/k/v/attn are [8192, 2048] row-major,
// head h occupies columns [h*256, h*256+256).
// LDS strides padded (260 / 20 / 18) for bank-conflict-free b64 gathers.
// ---------------------------------------------------------------------------
__global__ __launch_bounds__(32) void attn_flash_kernel(
    const float* __restrict__ q, const float* __restrict__ k,
    const float* __restrict__ v, float* __restrict__ attn) {
  __shared__ float qs[16 * 260];   // q tile   [row][256], stride 260
  __shared__ float vts[256 * 20];  // v tile^T [e][16],    stride 20
  __shared__ float ps[16 * 18];    // P tile   [row][16],  stride 18

  const int lane = threadIdx.x;
  const int r16 = lane & 15;
  const int hi = lane >> 4;
  const int klo = hi << 1;

  const int qt = blockIdx.x;  // query tile (16 rows)
  const int bh = blockIdx.y;
  const int b = bh >> 3;
  const int h = bh & 7;
  const size_t tq = (size_t)b * 1024 + (size_t)qt * 16;  // first query token
  const size_t hoff = (size_t)h * 256;

  // ---- stage q tile (16 x 256 fp32) into LDS --------------------------------
#pragma unroll 4
  for (int it = 0; it < 32; ++it) {
    int idx = it * 32 + lane;  // 1024 float4 total
    int row = idx >> 6;
    int c4 = idx & 63;
    float4 qv = *(const float4*)(q + (tq + row) * 2048 + hoff + c4 * 4);
    *(float4*)(&qs[row * 260 + c4 * 4]) = qv;
  }
  __syncthreads();

  v8f o[16];
  float mr[8], lr[8];
  {
    v8f zero = {0, 0, 0, 0, 0, 0, 0, 0};
#pragma unroll
    for (int t = 0; t < 16; ++t) o[t] = zero;
#pragma unroll
    for (int i = 0; i < 8; ++i) {
      mr[i] = -1.0e30f;  // finite "-inf": avoids inf-inf NaN on first tile
      lr[i] = 0.0f;
    }
  }

  for (int kt = 0; kt < 64; ++kt) {
    const size_t tk = (size_t)b * 1024 + (size_t)kt * 16;

    // prefetch next kv tile rows into cache (global_prefetch_b8)
    if (kt + 1 < 64) {
      const size_t tkn = tk + 16;
      __builtin_prefetch(k + (tkn + r16) * 2048 + hoff, 0, 1);
      __builtin_prefetch(v + (tkn + r16) * 2048 + hoff, 0, 1);
    }

    // ---- stage v tile transposed: vts[e][kv] --------------------------------
    // lane L: kv row = L&15, covers half of the 64 float4 of that row.
#pragma unroll 4
    for (int cc = 0; cc < 32; ++cc) {
      int c4 = hi * 32 + cc;
      float4 vv = *(const float4*)(v + (tk + r16) * 2048 + hoff + c4 * 4);
      vts[(c4 * 4 + 0) * 20 + r16] = vv.x;
      vts[(c4 * 4 + 1) * 20 + r16] = vv.y;
      vts[(c4 * 4 + 2) * 20 + r16] = vv.z;
      vts[(c4 * 4 + 3) * 20 + r16] = vv.w;
    }
    __syncthreads();

    // ---- S = q_tile @ k_tile^T : 64 WMMA (K = 256), 2-stage pipeline --------
    v8f s = {0, 0, 0, 0, 0, 0, 0, 0};
    {
      const float* krow = k + (tk + r16) * 2048 + hoff + klo;
      const float* qrow = &qs[r16 * 260 + klo];
      v2f aq0 = *(const v2f*)(qrow + 0);
      v2f bk0 = *(const v2f*)(krow + 0);
      v2f aq1, bk1;
      int ks = 0;
      for (; ks < 62; ks += 2) {
        aq1 = *(const v2f*)(qrow + (ks + 1) * 4);
        bk1 = *(const v2f*)(krow + (ks + 1) * 4);
        s = wmma_f32(aq0, bk0, s);
        aq0 = *(const v2f*)(qrow + (ks + 2) * 4);
        bk0 = *(const v2f*)(krow + (ks + 2) * 4);
        s = wmma_f32(aq1, bk1, s);
      }
      // ks == 62
      aq1 = *(const v2f*)(qrow + 63 * 4);
      bk1 = *(const v2f*)(krow + 63 * 4);
      s = wmma_f32(aq0, bk0, s);
      s = wmma_f32(aq1, bk1, s);
    }

    // ---- streaming softmax update (rows split across 16-lane halves) -------
    float alpha[8];
#pragma unroll
    for (int i = 0; i < 8; ++i) {
      float sv = s[i] * 0.0625f;  // 1/sqrt(256)
      float rm = sv;
      rm = fmaxf(rm, __shfl_xor(rm, 1, 32));
      rm = fmaxf(rm, __shfl_xor(rm, 2, 32));
      rm = fmaxf(rm, __shfl_xor(rm, 4, 32));
      rm = fmaxf(rm, __shfl_xor(rm, 8, 32));
      float mn = fmaxf(mr[i], rm);
      float al = __expf(mr[i] - mn);
      float p = __expf(sv - mn);
      float rs = p;
      rs += __shfl_xor(rs, 1, 32);
      rs += __shfl_xor(rs, 2, 32);
      rs += __shfl_xor(rs, 4, 32);
      rs += __shfl_xor(rs, 8, 32);
      lr[i] = lr[i] * al + rs;
      mr[i] = mn;
      alpha[i] = al;
      ps[(i + 8 * hi) * 18 + r16] = p;  // C-layout -> LDS bounce
    }
#pragma unroll
    for (int t = 0; t < 16; ++t) {
#pragma unroll
      for (int i = 0; i < 8; ++i) o[t][i] *= alpha[i];
    }
    __syncthreads();

    // ---- O += P(16x16) @ V(16x256) : 64 WMMA, rolling ds prefetch ----------
#pragma unroll
    for (int kc = 0; kc < 4; ++kc) {
      v2f a = *(const v2f*)(&ps[r16 * 18 + kc * 4 + klo]);
      v2f bv = *(const v2f*)(&vts[(0 * 16 + r16) * 20 + kc * 4 + klo]);
#pragma unroll
      for (int t = 0; t < 15; ++t) {
        v2f bvn = *(const v2f*)(&vts[((t + 1) * 16 + r16) * 20 + kc * 4 + klo]);
        o[t] = wmma_f32(a, bv, o[t]);
        bv = bvn;
      }
      o[15] = wmma_f32(a, bv, o[15]);
    }
    __syncthreads();
  }

  // ---- finalize: O / l, write to attn[token, h*256 + e] ---------------------
  float inv[8];
#pragma unroll
  for (int i = 0; i < 8; ++i) inv[i] = 1.0f / lr[i];
#pragma unroll
  for (int t = 0; t < 16; ++t) {
#pragma unroll
    for (int i = 0; i < 8; ++i) {
      attn[(tq + i + 8 * hi) * 2048 + hoff + t * 16 + r16] = o[t][i] * inv[i];
    }
  }
}

// ---------------------------------------------------------------------------
// Row LayerNorm over D=256. One 256-thread block per row (jnp population var).
// ---------------------------------------------------------------------------
__global__ __launch_bounds__(256) void layernorm256_kernel(
    const float* __restrict__ in, const float* __restrict__ g,
    const float* __restrict__ beta, float* __restrict__ out) {
  const int row = blockIdx.x;
  const int t = threadIdx.x;
  float x = in[(size_t)row * 256 + t];
  float s1 = x, s2 = x * x;
#pragma unroll
  for (int m = 1; m < 32; m <<= 1) {
    s1 += __shfl_xor(s1, m, 32);
    s2 += __shfl_xor(s2, m, 32);
  }
  __shared__ float r1[8], r2[8];
  if ((t & 31) == 0) {
    r1[t >> 5] = s1;
    r2[t >> 5] = s2;
  }
  __syncthreads();
  if (t == 0) {
    float a = 0.f, c = 0.f;
#pragma unroll
    for (int i = 0; i < 8; ++i) {
      a += r1[i];
      c += r2[i];
    }
    float mu = a * (1.0f / 256.0f);
    float var = c * (1.0f / 256.0f) - mu * mu;
    r1[0] = mu;
    r2[0] = rsqrtf(var + 1e-5f);
  }
  __syncthreads();
  float mu = r1[0], rstd = r2[0];
  out[(size_t)row * 256 + t] = (x - mu) * rstd * g[t] + beta[t];
}

// ---------------------------------------------------------------------------
// Orchestration. Workspace layout (floats):
//   q    [8192,2048]  @ 0
//   k    [8192,2048]  @ 16777216
//   v    [8192,2048]  @ 33554432
//   attn [8192,2048]  @ 50331648
//   y1   [8192, 256]  @ 67108864   (Wu proj + bu + x residual, pre-LN1)
//   x1   [8192, 256]  @ 69206016   (LN1 out)
//   hmid [8192,1024]  @ 71303168   (relu(x1 W1^T + b1))
//   z    [8192, 256]  @ 79691776   (hmid W2^T + b2 + x1, pre-LN2)
// total 81,788,928 floats = ~312 MiB
// ---------------------------------------------------------------------------
extern "C" void kernel_launch(void* const* d_in, const int* in_sizes, int n_in,
                              void* d_out, int out_size, void* d_ws,
                              size_t ws_size, hipStream_t stream) {
  (void)in_sizes;
  (void)n_in;
  (void)out_size;
  (void)ws_size;
  const float* x = (const float*)d_in[0];
  const float* Wq = (const float*)d_in[1];
  const float* Wk = (const float*)d_in[2];
  const float* Wv = (const float*)d_in[3];
  const float* Wu = (const float*)d_in[4];
  const float* bu = (const float*)d_in[5];
  const float* W1 = (const float*)d_in[6];
  const float* b1 = (const float*)d_in[7];
  const float* W2 = (const float*)d_in[8];
  const float* b2 = (const float*)d_in[9];
  const float* g1 = (const float*)d_in[10];
  const float* be1 = (const float*)d_in[11];
  const float* g2 = (const float*)d_in[12];
  const float* be2 = (const float*)d_in[13];
  float* out = (float*)d_out;

  float* ws = (float*)d_ws;
  float* q = ws + 0;
  float* k = ws + 16777216;
  float* v = ws + 33554432;
  float* attn = ws + 50331648;
  float* y1 = ws + 67108864;
  float* x1 = ws + 69206016;
  float* hmid = ws + 71303168;
  float* z = ws + 79691776;

  const int M = 8192;  // 8 * 1024 tokens

  // QKV projections: [8192,256] @ [2048,256]^T -> [8192,2048]
  gemm_wmma_f32_kernel<<<dim3(256, 8), 128, 0, stream>>>(x, Wq, nullptr,
                                                         nullptr, q, M, 2048,
                                                         256, 0);
  gemm_wmma_f32_kernel<<<dim3(256, 8), 128, 0, stream>>>(x, Wk, nullptr,
                                                         nullptr, k, M, 2048,
                                                         256, 0);
  gemm_wmma_f32_kernel<<<dim3(256, 8), 128, 0, stream>>>(x, Wv, nullptr,
                                                         nullptr, v, M, 2048,
                                                         256, 0);

  // Flash attention per (b,h): grid (64 q-tiles, 64 bh), 1 wave each
  attn_flash_kernel<<<dim3(64, 64), 32, 0, stream>>>(q, k, v, attn);

  // Output projection + bias + residual(x): [8192,2048] @ [256,2048]^T
  gemm_wmma_f32_kernel<<<dim3(256, 1), 128, 0, stream>>>(
      attn, Wu, bu, x, y1, M, 256, 2048, FLAG_BIAS | FLAG_RES);

  // LN1
  layernorm256_kernel<<<8192, 256, 0, stream>>>(y1, g1, be1, x1);

  // MLP up + relu: [8192,256] @ [1024,256]^T
  gemm_wmma_f32_kernel<<<dim3(256, 4), 128, 0, stream>>>(
      x1, W1, b1, nullptr, hmid, M, 1024, 256, FLAG_BIAS | FLAG_RELU);

  // MLP down + bias + residual(x1): [8192,1024] @ [256,1024]^T
  gemm_wmma_f32_kernel<<<dim3(256, 1), 128, 0, stream>>>(
      hmid, W2, b2, x1, z, M, 256, 1024, FLAG_BIAS | FLAG_RES);

  // LN2 -> out
  layernorm256_kernel<<<8192, 256, 0, stream>>>(z, g2, be2, out);
}